// VFR_SM_4964982194348
// MI455X (gfx1250) — compile-verified
//
#include <hip/hip_runtime.h>
#include <hip/hip_bf16.h>

typedef float v2f __attribute__((ext_vector_type(2)));
typedef float v4f __attribute__((ext_vector_type(4)));
typedef float v8f __attribute__((ext_vector_type(8)));

#define B_  4
#define D_  64
#define F_  64
#define E_  16
#define SCALE (1.0f / 4096.0f)

// ---------------------------------------------------------------------------
// Kernel 1: streaming pooling pass. One pass over x producing
//   sum_c[b,c,f] (sum over a,s), sum_s[b,s,f] (sum over a,c), sum_a[b,a,f]
// Grid: B*A*4 blocks (each owns 16 s-rows of one (b,a) slab), 256 threads.
// Thread t: f = t&63, quarter q = t>>6 owns c in [q*16, q*16+16).
// ---------------------------------------------------------------------------
__global__ void se_pool_kernel(const float* __restrict__ x,
                               float* __restrict__ sum_c,
                               float* __restrict__ sum_s,
                               float* __restrict__ sum_a) {
    const int blk = blockIdx.x;
    const int sc  = blk & 3;            // s-chunk (16 rows)
    const int a   = (blk >> 2) & 63;
    const int b   = blk >> 8;

    const int t = threadIdx.x;
    const int f = t & 63;
    const int q = t >> 6;

    const float* xb = x + ((size_t)((b * 64 + a) * 64 + sc * 16)) * 4096;

    float accc[16];
#pragma unroll
    for (int i = 0; i < 16; ++i) accc[i] = 0.0f;
    float acca = 0.0f;

    __shared__ float red[256];

    for (int si = 0; si < 16; ++si) {
        const int s = sc * 16 + si;
        const float* xr = xb + (size_t)si * 4096 + (q * 16) * 64 + f;
        float row = 0.0f;
#pragma unroll
        for (int i = 0; i < 16; ++i) {
            float v = __builtin_nontemporal_load(xr + i * 64);
            accc[i] += v;
            row += v;
        }
        acca += row;
        red[t] = row;
        __syncthreads();
        if (q == 0) {
            float tot = red[f] + red[64 + f] + red[128 + f] + red[192 + f];
            atomicAdd(&sum_s[(b * 64 + s) * 64 + f], tot);
        }
        __syncthreads();
    }

    red[t] = acca;
    __syncthreads();
    if (q == 0) {
        float tot = red[f] + red[64 + f] + red[128 + f] + red[192 + f];
        atomicAdd(&sum_a[(b * 64 + a) * 64 + f], tot);
    }

#pragma unroll
    for (int i = 0; i < 16; ++i)
        atomicAdd(&sum_c[(b * 64 + (q * 16 + i)) * 64 + f], accc[i]);
}

// ---------------------------------------------------------------------------
// Kernel 2: per-filter tri-axis SE bottleneck via V_WMMA_F32_16X16X4_F32.
// 192 problems (3 axes x 64 filters), one wave32 per problem.
//   z[4x16]  = relu( (sum/4096)[4x64] @ w1[f][64x16] )   -> 16 wmma k-steps
//   att[4x64]= sigmoid( z[4x16] @ w2[f][16x64] )         -> 4 tiles x 4 steps
// A-frag layout (16x4 f32): lane<16 -> (M=lane, K={0,1}); lane>=16 -> K={2,3}
// B-frag layout (4x16 f32): mirrored with N in place of M.
// C/D layout (16x16 f32):  vgpr r, lane<16 -> (M=r, N=lane).
// ---------------------------------------------------------------------------
__global__ void se_attn_kernel(const float* __restrict__ sums,   // 3 x [B,64,F]
                               float* __restrict__ atts,         // 3 x [B,64,F]
                               const float* __restrict__ w1_a,
                               const float* __restrict__ w1_s,
                               const float* __restrict__ w1_c,
                               const float* __restrict__ w2_a,
                               const float* __restrict__ w2_s,
                               const float* __restrict__ w2_c) {
    const int lane = threadIdx.x & 31;
    const int wid  = threadIdx.x >> 5;
    const int p    = blockIdx.x * 8 + wid;    // 0..191
    const int axis = p >> 6;                  // 0=c, 1=s, 2=a
    const int f    = p & 63;

    const float* msum = sums + axis * (B_ * D_ * F_);
    float*       att  = atts + axis * (B_ * D_ * F_);
    const float* w1 = (axis == 0) ? w1_c : (axis == 1) ? w1_s : w1_a; // [F][64][16]
    const float* w2 = (axis == 0) ? w2_c : (axis == 1) ? w2_s : w2_a; // [F][16][64]

    const int Mrow  = lane & 15;
    const int khalf = lane >> 4;       // which K pair this lane holds
    const int mr    = Mrow & 3;        // clamped row for safe loads
    const float msk = (Mrow < 4) ? 1.0f : 0.0f;

    __shared__ float zbuf[8][4][16];

#if __has_builtin(__builtin_amdgcn_wmma_f32_16x16x4_f32)
    // ---- GEMM1: z = m @ w1 ------------------------------------------------
    v8f z = {0.f, 0.f, 0.f, 0.f, 0.f, 0.f, 0.f, 0.f};
#pragma unroll
    for (int k0 = 0; k0 < 16; ++k0) {
        const int kb = k0 * 4 + khalf * 2;
        v2f afrag, bfrag;
        afrag.x = msum[(mr * 64 + kb)     * 64 + f] * (SCALE * msk);
        afrag.y = msum[(mr * 64 + kb + 1) * 64 + f] * (SCALE * msk);
        bfrag.x = w1[(f * 64 + kb)     * 16 + Mrow];
        bfrag.y = w1[(f * 64 + kb + 1) * 16 + Mrow];
        z = __builtin_amdgcn_wmma_f32_16x16x4_f32(false, afrag, false, bfrag,
                                                  (short)0, z, false, false);
    }
    // relu + stash the 4 valid rows of z for A-fragment re-layout
    if (lane < 16) {
        zbuf[wid][0][lane] = fmaxf(z[0], 0.0f);
        zbuf[wid][1][lane] = fmaxf(z[1], 0.0f);
        zbuf[wid][2][lane] = fmaxf(z[2], 0.0f);
        zbuf[wid][3][lane] = fmaxf(z[3], 0.0f);
    }
    __syncthreads();

    // ---- GEMM2: att = sigmoid(z @ w2) ------------------------------------
#pragma unroll
    for (int j = 0; j < 4; ++j) {                 // N tiles over d-out
        v8f y = {0.f, 0.f, 0.f, 0.f, 0.f, 0.f, 0.f, 0.f};
#pragma unroll
        for (int k1 = 0; k1 < 4; ++k1) {
            const int eb = k1 * 4 + khalf * 2;
            v2f afrag, bfrag;
            afrag.x = zbuf[wid][mr][eb]     * msk;
            afrag.y = zbuf[wid][mr][eb + 1] * msk;
            const int n = j * 16 + Mrow;
            bfrag.x = w2[(f * 16 + eb)     * 64 + n];
            bfrag.y = w2[(f * 16 + eb + 1) * 64 + n];
            y = __builtin_amdgcn_wmma_f32_16x16x4_f32(false, afrag, false, bfrag,
                                                      (short)0, y, false, false);
        }
        if (lane < 16) {
#pragma unroll
            for (int r = 0; r < 4; ++r) {
                float v  = y[r];
                float sg = 1.0f / (1.0f + __expf(-v));
                att[(r * 64 + (j * 16 + lane)) * 64 + f] = sg;
            }
        }
    }
#else
    // Scalar fallback (keeps compile green if the f32 WMMA builtin is absent)
    if (lane < 16) {
        const int e = lane;
#pragma unroll
        for (int b = 0; b < 4; ++b) {
            float acc = 0.0f;
            for (int d = 0; d < 64; ++d)
                acc += msum[(b * 64 + d) * 64 + f] * SCALE * w1[(f * 64 + d) * 16 + e];
            zbuf[wid][b][e] = fmaxf(acc, 0.0f);
        }
    }
    __syncthreads();
#pragma unroll
    for (int half = 0; half < 2; ++half) {
        const int d = half * 32 + lane;
#pragma unroll
        for (int b = 0; b < 4; ++b) {
            float acc = 0.0f;
            for (int e = 0; e < 16; ++e)
                acc += zbuf[wid][b][e] * w2[(f * 16 + e) * 64 + d];
            att[(b * 64 + d) * 64 + f] = 1.0f / (1.0f + __expf(-acc));
        }
    }
#endif
}

// ---------------------------------------------------------------------------
// Kernel 3: out = x * att_c * att_s * att_a  (streaming, float4, NT)
// Grid: B*A*S = 16384 blocks, 256 threads; each block one [C=64,F=64] slab.
// Thread t: f-chunk fq = t&15 (4 floats), c rows c0 = t>>4 (+16 stride).
// ---------------------------------------------------------------------------
__global__ void se_apply_kernel(const float* __restrict__ x,
                                const float* __restrict__ att_c,
                                const float* __restrict__ att_s,
                                const float* __restrict__ att_a,
                                float* __restrict__ out) {
    const int blk = blockIdx.x;
    const int s = blk & 63;
    const int a = (blk >> 6) & 63;
    const int b = blk >> 12;

    const v4f* xs = (const v4f*)(x   + (size_t)blk * 4096);
    v4f*       os = (v4f*)      (out + (size_t)blk * 4096);

    const int t  = threadIdx.x;
    const int fq = t & 15;     // float4 chunk along f
    const int c0 = t >> 4;     // base c row

    const v4f sa = *(const v4f*)(att_s + ((b * 64 + s) * 64) + fq * 4);
    const v4f aa = *(const v4f*)(att_a + ((b * 64 + a) * 64) + fq * 4);
    const v4f m  = sa * aa;

#pragma unroll
    for (int i = 0; i < 4; ++i) {
        const int c = c0 + i * 16;
        v4f xv = __builtin_nontemporal_load(&xs[c * 16 + fq]);
        v4f cc = *(const v4f*)(att_c + ((b * 64 + c) * 64) + fq * 4);
        __builtin_nontemporal_store(xv * m * cc, &os[c * 16 + fq]);
    }
}

// ---------------------------------------------------------------------------
extern "C" void kernel_launch(void* const* d_in, const int* in_sizes, int n_in,
                              void* d_out, int out_size, void* d_ws, size_t ws_size,
                              hipStream_t stream) {
    const float* x    = (const float*)d_in[0];
    const float* w1_a = (const float*)d_in[1];
    const float* w1_s = (const float*)d_in[2];
    const float* w1_c = (const float*)d_in[3];
    const float* w2_a = (const float*)d_in[4];
    const float* w2_s = (const float*)d_in[5];
    const float* w2_c = (const float*)d_in[6];
    float* out = (float*)d_out;

    const int SEG = B_ * D_ * F_;              // 16384 floats
    float* wsf   = (float*)d_ws;
    float* sum_c = wsf;                        // [B,C,F]
    float* sum_s = wsf + SEG;                  // [B,S,F]
    float* sum_a = wsf + 2 * SEG;              // [B,A,F]
    float* atts  = wsf + 3 * SEG;              // att_c, att_s, att_a

    // zero the atomic accumulation buffers (graph-capture safe memset node)
    hipMemsetAsync(d_ws, 0, (size_t)(3 * SEG) * sizeof(float), stream);

    se_pool_kernel<<<B_ * 64 * 4, 256, 0, stream>>>(x, sum_c, sum_s, sum_a);
    se_attn_kernel<<<24, 256, 0, stream>>>(wsf, atts,
                                           w1_a, w1_s, w1_c, w2_a, w2_s, w2_c);
    se_apply_kernel<<<B_ * 64 * 64, 256, 0, stream>>>(
        x, atts, atts + SEG, atts + 2 * SEG, out);
}